// TuckerMoELayer_72370198937819
// MI455X (gfx1250) — compile-verified
//
#include <hip/hip_runtime.h>
#include <math.h>

// ---------------- problem constants ----------------
#define TT   1024   // tokens
#define DD   1024   // d_model
#define DFF  1024   // expert intermediate
#define RR   64     // tucker rank
#define GG   8      // groups
#define EE   32     // experts per group
#define NEXP 256    // total experts
#define KK   8      // top-k
#define FF   (TT*KK) // 8192 pairs

// LDS A-tile: 16 rows x 32 k, row stride 36 words (36 even -> 8B-aligned v2f
// loads; 36 = 4*9 with 9 coprime to 16 -> the 16 row addresses land in 16
// distinct banks of the 64-bank LDS)
#define ASTRIDE 36

typedef float v2f __attribute__((ext_vector_type(2)));
typedef float v8f __attribute__((ext_vector_type(8)));

static __device__ __forceinline__ v8f wmma_f32(v2f a, v2f b, v8f c) {
  // V_WMMA_F32_16X16X4_F32 : D = A(16x4) * B(4x16) + C(16x16), fp32
  return __builtin_amdgcn_wmma_f32_16x16x4_f32(false, a, false, b, (short)0, c,
                                               false, false);
}

// ---------------------------------------------------------------------------
// K1: router logits = x @ Wg   [1024x1024]x[1024x256]
// grid (T/16, NEXP/64), block 128 (4 waves, 16 cols each).
// A tile staged in LDS once per block (shared by all 4 waves).
// ---------------------------------------------------------------------------
__global__ __launch_bounds__(128) void k_router(const float* __restrict__ x,
                                                const float* __restrict__ Wg,
                                                float* __restrict__ logits) {
  __shared__ float a_s[16 * ASTRIDE];
  const int lane = threadIdx.x & 31, wv = threadIdx.x >> 5;
  const int l = lane & 15, h = lane >> 4;
  const int col = blockIdx.y * 64 + wv * 16 + l;
  const float* bp = Wg + (size_t)(2 * h) * NEXP + col;
  // cooperative A loader: 8 threads per row, float4 each
  const int lrow = threadIdx.x >> 3;
  const int lko  = (threadIdx.x & 7) * 4;
  const float* xrow = x + (size_t)(blockIdx.x * 16 + lrow) * DD + lko;
  v8f acc = {};
  for (int kc = 0; kc < DD; kc += 32) {
    *(float4*)&a_s[lrow * ASTRIDE + lko] = *(const float4*)(xrow + kc);
    if (kc + 32 < DD)
      __builtin_prefetch(bp + (size_t)(kc + 32) * NEXP, 0, 1);
    __syncthreads();
#pragma unroll
    for (int k = 0; k < 32; k += 4) {
      v2f a = *(const v2f*)&a_s[l * ASTRIDE + k + 2 * h];
      v2f b = {bp[(size_t)(kc + k) * NEXP], bp[(size_t)(kc + k + 1) * NEXP]};
      acc = wmma_f32(a, b, acc);
    }
    __syncthreads();
  }
#pragma unroll
  for (int j = 0; j < 8; ++j)
    logits[(size_t)(blockIdx.x * 16 + j + 8 * h) * NEXP + col] = acc[j];
}

// ---------------------------------------------------------------------------
// K2: per-token top-8 (first-index tie-break, like jax top_k) + softmax
// one wave per token, 8 tokens / block
// ---------------------------------------------------------------------------
__global__ __launch_bounds__(256) void k_topk(const float* __restrict__ logits,
                                              float* __restrict__ wtop,
                                              int* __restrict__ seltop) {
  const int lane = threadIdx.x & 31, wv = threadIdx.x >> 5;
  const int t = blockIdx.x * 8 + wv;
  float v[8];
#pragma unroll
  for (int j = 0; j < 8; ++j) v[j] = logits[(size_t)t * NEXP + j * 32 + lane];
  __shared__ float sv[8][KK];
  __shared__ int   si[8][KK];
  for (int it = 0; it < KK; ++it) {
    float bv = v[0];
    int   bi = lane;
#pragma unroll
    for (int j = 1; j < 8; ++j) {
      int idx = j * 32 + lane;
      if (v[j] > bv || (v[j] == bv && idx < bi)) { bv = v[j]; bi = idx; }
    }
#pragma unroll
    for (int off = 16; off > 0; off >>= 1) {
      float ov = __shfl_xor(bv, off, 32);
      int   oi = __shfl_xor(bi, off, 32);
      if (ov > bv || (ov == bv && oi < bi)) { bv = ov; bi = oi; }
    }
    if (lane == 0) { sv[wv][it] = bv; si[wv][it] = bi; }
#pragma unroll
    for (int j = 0; j < 8; ++j)
      if ((bi >> 5) == j && (bi & 31) == lane) v[j] = -3.4e38f;
  }
  if (lane < KK) {
    float m = sv[wv][0], s = 0.f;
#pragma unroll
    for (int j = 0; j < KK; ++j) s += expf(sv[wv][j] - m);
    wtop[(size_t)t * KK + lane]   = expf(sv[wv][lane] - m) / s;
    seltop[(size_t)t * KK + lane] = si[wv][lane];
  }
}

// ---------------------------------------------------------------------------
// K3: pre[g,t,r] = x @ uin_{gate,up}[g]  for all groups (both in one pass)
// grid (T/16, G), block 128 (4 waves cover r=0..63). LDS-staged A tile.
// ---------------------------------------------------------------------------
__global__ __launch_bounds__(128) void k_prein(const float* __restrict__ x,
                                               const float* __restrict__ uing,
                                               const float* __restrict__ uinu,
                                               float* __restrict__ preg,
                                               float* __restrict__ preu) {
  __shared__ float a_s[16 * ASTRIDE];
  const int g = blockIdx.y;
  const int lane = threadIdx.x & 31, wv = threadIdx.x >> 5;
  const int l = lane & 15, h = lane >> 4;
  const int col = wv * 16 + l;
  const float* bg = uing + (size_t)g * DD * RR + (size_t)(2 * h) * RR + col;
  const float* bu = uinu + (size_t)g * DD * RR + (size_t)(2 * h) * RR + col;
  const int lrow = threadIdx.x >> 3;
  const int lko  = (threadIdx.x & 7) * 4;
  const float* xrow = x + (size_t)(blockIdx.x * 16 + lrow) * DD + lko;
  v8f cg = {}, cu = {};
  for (int kc = 0; kc < DD; kc += 32) {
    *(float4*)&a_s[lrow * ASTRIDE + lko] = *(const float4*)(xrow + kc);
    if (kc + 32 < DD) {
      __builtin_prefetch(bg + (size_t)(kc + 32) * RR, 0, 1);
      __builtin_prefetch(bu + (size_t)(kc + 32) * RR, 0, 1);
    }
    __syncthreads();
#pragma unroll
    for (int k = 0; k < 32; k += 4) {
      v2f a  = *(const v2f*)&a_s[l * ASTRIDE + k + 2 * h];
      v2f b1 = {bg[(size_t)(kc + k) * RR], bg[(size_t)(kc + k + 1) * RR]};
      cg = wmma_f32(a, b1, cg);
      v2f b2 = {bu[(size_t)(kc + k) * RR], bu[(size_t)(kc + k + 1) * RR]};
      cu = wmma_f32(a, b2, cu);
    }
    __syncthreads();
  }
#pragma unroll
  for (int j = 0; j < 8; ++j) {
    size_t o = (size_t)g * TT * RR + (size_t)(blockIdx.x * 16 + j + 8 * h) * RR + col;
    preg[o] = cg[j];
    preu[o] = cu[j];
  }
}

// ---------------------------------------------------------------------------
// K4: deterministic counting sort of the 8192 (token,expert) pairs by group.
// single block, stable rank via LDS scan.
// ---------------------------------------------------------------------------
__global__ __launch_bounds__(256) void k_bucket(const int* __restrict__ seltop,
                                                const float* __restrict__ wtop,
                                                int* __restrict__ gs,
                                                int* __restrict__ srt_tok,
                                                int* __restrict__ srt_e,
                                                float* __restrict__ srt_w) {
  __shared__ int hist[GG];
  __shared__ int cursor[GG];
  __shared__ int sg[256];
  const int tid = threadIdx.x;
  if (tid < GG) hist[tid] = 0;
  __syncthreads();
  for (int f = tid; f < FF; f += 256) atomicAdd(&hist[seltop[f] >> 5], 1);
  __syncthreads();
  if (tid == 0) {
    int acc = 0;
    for (int g = 0; g < GG; ++g) { gs[g] = acc; cursor[g] = acc; acc += hist[g]; }
    gs[GG] = acc;
  }
  __syncthreads();
  for (int c = 0; c < FF; c += 256) {
    const int f = c + tid;
    const int e = seltop[f];
    const int g = e >> 5;
    sg[tid] = g;
    __syncthreads();
    int rank = 0;
    for (int j = 0; j < tid; ++j) rank += (sg[j] == g);
    const int pos = cursor[g] + rank;
    srt_tok[pos] = f >> 3;  // K = 8
    srt_e[pos]   = e;
    srt_w[pos]   = wtop[f];
    __syncthreads();
    if (tid < GG) {
      int cc = 0;
      for (int j = 0; j < 256; ++j) cc += (sg[j] == tid);
      cursor[tid] += cc;
    }
    __syncthreads();
  }
}

// ---------------------------------------------------------------------------
// K5: per-pair Tucker core bmm, input gathered from pre[g,tok,:] (wave/pair)
// ---------------------------------------------------------------------------
__global__ __launch_bounds__(256) void k_core_gather(const float* __restrict__ pre,
                                                     const float* __restrict__ core,
                                                     const int* __restrict__ srt_tok,
                                                     const int* __restrict__ srt_e,
                                                     float* __restrict__ hout) {
  const int lane = threadIdx.x & 31, wv = threadIdx.x >> 5;
  const int p = blockIdx.x * 8 + wv;
  const int e = srt_e[p], g = e >> 5;
  const float* xin = pre + (size_t)g * TT * RR + (size_t)srt_tok[p] * RR;
  const float* cs  = core + (size_t)e * RR * RR;
  float a0 = 0.f, a1 = 0.f;
#pragma unroll 4
  for (int k = 0; k < RR; ++k) {
    float xv = xin[k];
    a0 = fmaf(xv, cs[(size_t)k * RR + lane], a0);
    a1 = fmaf(xv, cs[(size_t)k * RR + lane + 32], a1);
  }
  hout[(size_t)p * RR + lane]      = a0;
  hout[(size_t)p * RR + lane + 32] = a1;
}

// K8 variant: input already in sorted pair order [F][R]
__global__ __launch_bounds__(256) void k_core_sorted(const float* __restrict__ xin_s,
                                                     const float* __restrict__ core,
                                                     const int* __restrict__ srt_e,
                                                     float* __restrict__ hout) {
  const int lane = threadIdx.x & 31, wv = threadIdx.x >> 5;
  const int p = blockIdx.x * 8 + wv;
  const int e = srt_e[p];
  const float* xin = xin_s + (size_t)p * RR;
  const float* cs  = core + (size_t)e * RR * RR;
  float a0 = 0.f, a1 = 0.f;
#pragma unroll 4
  for (int k = 0; k < RR; ++k) {
    float xv = xin[k];
    a0 = fmaf(xv, cs[(size_t)k * RR + lane], a0);
    a1 = fmaf(xv, cs[(size_t)k * RR + lane + 32], a1);
  }
  hout[(size_t)p * RR + lane]      = a0;
  hout[(size_t)p * RR + lane + 32] = a1;
}

// ---------------------------------------------------------------------------
// K6: bucketed GEMM [cnt x 64] @ uout_{gate,up}[g] (64 x 1024) + fused SwiGLU
// grid (F/16, DFF/64, G) with uniform early-exit on bucket size
// ---------------------------------------------------------------------------
__global__ __launch_bounds__(128) void k_gateup(const float* __restrict__ hg,
                                                const float* __restrict__ hu,
                                                const float* __restrict__ uog,
                                                const float* __restrict__ uou,
                                                const int* __restrict__ gs,
                                                float* __restrict__ act) {
  const int g = blockIdx.z;
  const int gstart = gs[g];
  const int cnt = gs[g + 1] - gstart;
  const int rt = blockIdx.x;
  if (rt * 16 >= cnt) return;
  const int lane = threadIdx.x & 31, wv = threadIdx.x >> 5;
  const int l = lane & 15, h = lane >> 4;
  int arow = rt * 16 + l;
  if (arow >= cnt) arow = cnt - 1;  // clamp partial tile (keeps EXEC full)
  const int pa  = gstart + arow;
  const int col = blockIdx.y * 64 + wv * 16 + l;
  const float* ag = hg + (size_t)pa * RR + 2 * h;
  const float* au = hu + (size_t)pa * RR + 2 * h;
  const float* bg = uog + (size_t)g * RR * DFF + (size_t)(2 * h) * DFF + col;
  const float* bu = uou + (size_t)g * RR * DFF + (size_t)(2 * h) * DFF + col;
  v8f cg = {}, cu = {};
#pragma unroll
  for (int k = 0; k < RR; k += 4) {
    v2f a1 = {ag[k], ag[k + 1]};
    v2f b1 = {bg[(size_t)k * DFF], bg[(size_t)(k + 1) * DFF]};
    cg = wmma_f32(a1, b1, cg);
    v2f a2 = {au[k], au[k + 1]};
    v2f b2 = {bu[(size_t)k * DFF], bu[(size_t)(k + 1) * DFF]};
    cu = wmma_f32(a2, b2, cu);
  }
#pragma unroll
  for (int j = 0; j < 8; ++j) {
    const int rl = rt * 16 + j + 8 * h;
    if (rl < cnt) {
      float gv = cg[j];
      float s  = gv / (1.f + expf(-gv));  // silu
      act[(size_t)(gstart + rl) * DFF + col] = s * cu[j];
    }
  }
}

// ---------------------------------------------------------------------------
// K7: bucketed GEMM [cnt x 1024] @ uin_down[g] (1024 x 64) -> xin_d [F x 64]
// grid (F/16, G), block 128. LDS-staged A tile (act rows).
// ---------------------------------------------------------------------------
__global__ __launch_bounds__(128) void k_downin(const float* __restrict__ act,
                                                const float* __restrict__ uind,
                                                const int* __restrict__ gs,
                                                float* __restrict__ xind) {
  __shared__ float a_s[16 * ASTRIDE];
  const int g = blockIdx.y;
  const int gstart = gs[g];
  const int cnt = gs[g + 1] - gstart;
  const int rt = blockIdx.x;
  if (rt * 16 >= cnt) return;
  const int lane = threadIdx.x & 31, wv = threadIdx.x >> 5;
  const int l = lane & 15, h = lane >> 4;
  const int col = wv * 16 + l;
  const float* bp = uind + (size_t)g * DFF * RR + (size_t)(2 * h) * RR + col;
  // cooperative A loader with clamped (duplicated) tail rows
  const int lrow = threadIdx.x >> 3;
  const int lko  = (threadIdx.x & 7) * 4;
  int ldr = rt * 16 + lrow;
  if (ldr >= cnt) ldr = cnt - 1;
  const float* arow = act + (size_t)(gstart + ldr) * DFF + lko;
  v8f acc = {};
  for (int kc = 0; kc < DFF; kc += 32) {
    *(float4*)&a_s[lrow * ASTRIDE + lko] = *(const float4*)(arow + kc);
    if (kc + 32 < DFF)
      __builtin_prefetch(bp + (size_t)(kc + 32) * RR, 0, 1);
    __syncthreads();
#pragma unroll
    for (int k = 0; k < 32; k += 4) {
      v2f a = *(const v2f*)&a_s[l * ASTRIDE + k + 2 * h];
      v2f b = {bp[(size_t)(kc + k) * RR], bp[(size_t)(kc + k + 1) * RR]};
      acc = wmma_f32(a, b, acc);
    }
    __syncthreads();
  }
#pragma unroll
  for (int j = 0; j < 8; ++j) {
    const int rl = rt * 16 + j + 8 * h;
    if (rl < cnt) xind[(size_t)(gstart + rl) * RR + col] = acc[j];
  }
}

// ---------------------------------------------------------------------------
// K9: bucketed GEMM [cnt x 64] @ uout_down[g] (64 x 1024), scale by routing
// weight, fp32 atomic scatter-add into out[token, :]
// ---------------------------------------------------------------------------
__global__ __launch_bounds__(128) void k_downout(const float* __restrict__ hd,
                                                 const float* __restrict__ uod,
                                                 const int* __restrict__ gs,
                                                 const int* __restrict__ srt_tok,
                                                 const float* __restrict__ srt_w,
                                                 float* __restrict__ out) {
  const int g = blockIdx.z;
  const int gstart = gs[g];
  const int cnt = gs[g + 1] - gstart;
  const int rt = blockIdx.x;
  if (rt * 16 >= cnt) return;
  const int lane = threadIdx.x & 31, wv = threadIdx.x >> 5;
  const int l = lane & 15, h = lane >> 4;
  int arow = rt * 16 + l;
  if (arow >= cnt) arow = cnt - 1;
  const int pa  = gstart + arow;
  const int col = blockIdx.y * 64 + wv * 16 + l;
  const float* ap = hd + (size_t)pa * RR + 2 * h;
  const float* bp = uod + (size_t)g * RR * DD + (size_t)(2 * h) * DD + col;
  v8f acc = {};
#pragma unroll
  for (int k = 0; k < RR; k += 4) {
    v2f a = {ap[k], ap[k + 1]};
    v2f b = {bp[(size_t)k * DD], bp[(size_t)(k + 1) * DD]};
    acc = wmma_f32(a, b, acc);
  }
#pragma unroll
  for (int j = 0; j < 8; ++j) {
    const int rl = rt * 16 + j + 8 * h;
    if (rl < cnt) {
      const int p = gstart + rl;
      atomicAdd(&out[(size_t)srt_tok[p] * DD + col], acc[j] * srt_w[p]);
    }
  }
}

// ---------------------------------------------------------------------------
extern "C" void kernel_launch(void* const* d_in, const int* in_sizes, int n_in,
                              void* d_out, int out_size, void* d_ws, size_t ws_size,
                              hipStream_t stream) {
  (void)in_sizes; (void)n_in; (void)out_size; (void)ws_size;
  const float* x    = (const float*)d_in[0];
  const float* Wg   = (const float*)d_in[1];
  const float* uing = (const float*)d_in[2];
  const float* cg   = (const float*)d_in[3];
  const float* uog  = (const float*)d_in[4];
  const float* uinu = (const float*)d_in[5];
  const float* cu   = (const float*)d_in[6];
  const float* uou  = (const float*)d_in[7];
  const float* uind = (const float*)d_in[8];
  const float* cd   = (const float*)d_in[9];
  const float* uod  = (const float*)d_in[10];
  float* out = (float*)d_out;

  // workspace layout (float elements)
  float* WS      = (float*)d_ws;
  float* logits  = WS;                    // 262144
  float* wtop    = WS + 262144;           // 8192
  float* preg    = WS + 270336;           // 524288
  float* preu    = WS + 794624;           // 524288
  float* hg      = WS + 1318912;          // 524288
  float* hu      = WS + 1843200;          // 524288
  float* act     = WS + 2367488;          // 8388608
  float* xind    = WS + 10756096;         // 524288
  float* hd      = WS + 11280384;         // 524288
  float* srt_w   = WS + 11804672;         // 8192
  int*   seltop  = (int*)(WS + 11812864); // 8192
  int*   gs      = (int*)(WS + 11821056); // 16
  int*   srt_tok = (int*)(WS + 11821072); // 8192
  int*   srt_e   = (int*)(WS + 11829264); // 8192

  hipMemsetAsync(out, 0, (size_t)TT * DD * sizeof(float), stream);

  k_router<<<dim3(TT / 16, NEXP / 64), 128, 0, stream>>>(x, Wg, logits);
  k_topk<<<TT / 8, 256, 0, stream>>>(logits, wtop, seltop);
  k_prein<<<dim3(TT / 16, GG), 128, 0, stream>>>(x, uing, uinu, preg, preu);
  k_bucket<<<1, 256, 0, stream>>>(seltop, wtop, gs, srt_tok, srt_e, srt_w);
  k_core_gather<<<FF / 8, 256, 0, stream>>>(preg, cg, srt_tok, srt_e, hg);
  k_core_gather<<<FF / 8, 256, 0, stream>>>(preu, cu, srt_tok, srt_e, hu);
  k_gateup<<<dim3(FF / 16, DFF / 64, GG), 128, 0, stream>>>(hg, hu, uog, uou, gs, act);
  k_downin<<<dim3(FF / 16, GG), 128, 0, stream>>>(act, uind, gs, xind);
  k_core_sorted<<<FF / 8, 256, 0, stream>>>(xind, cd, srt_e, hd);
  k_downout<<<dim3(FF / 16, DD / 64, GG), 128, 0, stream>>>(hd, uod, gs, srt_tok, srt_w, out);
}